// LOTSinkhorn_37048387895361
// MI455X (gfx1250) — compile-verified
//
#include <hip/hip_runtime.h>
#include <cmath>

// Shapes fixed by the reference setup_inputs()
#define BB   64
#define NN   2048
#define DD   128
#define NR   256

typedef float v2f __attribute__((ext_vector_type(2)));
typedef float v8f __attribute__((ext_vector_type(8)));

// ---------------------------------------------------------------------------
// Kernel 1: squared norms of reference rows (256) and X rows (B*N), wave/row.
// ---------------------------------------------------------------------------
__global__ void sqnorm_kernel(const float* __restrict__ X,
                              const float* __restrict__ ref,
                              float* __restrict__ r2,
                              float* __restrict__ x2) {
    int wid  = (blockIdx.x * blockDim.x + threadIdx.x) >> 5;
    int lane = threadIdx.x & 31;
    const float* src;
    float* dst;
    if (wid < NR) {
        src = ref + (size_t)wid * DD;
        dst = r2 + wid;
    } else {
        size_t row = (size_t)wid - NR;          // 0 .. B*N-1
        src = X + row * DD;
        dst = x2 + row;
    }
    float s = 0.0f;
    #pragma unroll
    for (int k = 0; k < DD; k += 32) {
        float t = src[k + lane];
        s += t * t;
    }
    #pragma unroll
    for (int off = 16; off; off >>= 1) s += __shfl_xor(s, off, 32);
    if (lane == 0) *dst = s;
}

// ---------------------------------------------------------------------------
// Kernel 2: zero the v vector (first Sinkhorn iteration reads v = 0).
// ---------------------------------------------------------------------------
__global__ void zerov_kernel(float* __restrict__ v) {
    v[blockIdx.x * blockDim.x + threadIdx.x] = 0.0f;
}

// ---------------------------------------------------------------------------
// Kernel 3: C[b,i,j] = sqrt(max(|ref_i|^2 + |x_bj|^2 - 2*ref_i.x_bj, 0))
// Gram term via V_WMMA_F32_16X16X4_F32; one wave per 16x16 tile of C.
// A-layout (16x4 f32): lane m=L%16, VGPR0 holds K=khalf, VGPR1 K=khalf+1,
// khalf = (L>=16)?2:0.  B-layout (4x16) symmetric.  C/D: VGPR r -> M=r(+8).
// ---------------------------------------------------------------------------
__global__ void gram_kernel(const float* __restrict__ X,
                            const float* __restrict__ ref,
                            const float* __restrict__ r2,
                            const float* __restrict__ x2,
                            float* __restrict__ C) {
    int wid  = (blockIdx.x * blockDim.x + threadIdx.x) >> 5;
    int lane = threadIdx.x & 31;
    int b    = wid >> 11;            // / (16 * 128) tiles
    int rem  = wid & 2047;
    int i0   = (rem >> 7) * 16;
    int j0   = (rem & 127) * 16;

    int m     = lane & 15;
    int khalf = (lane >> 4) << 1;    // 0 or 2

    const float* Arow = ref + (size_t)(i0 + m) * DD + khalf;
    const float* Brow = X + ((size_t)b * NN + (j0 + m)) * DD + khalf;

    v8f acc = {};
    #pragma unroll 4
    for (int k = 0; k < DD; k += 4) {
        v2f a, bb;
        a.x  = Arow[k];     a.y  = Arow[k + 1];
        bb.x = Brow[k];     bb.y = Brow[k + 1];
        acc = __builtin_amdgcn_wmma_f32_16x16x4_f32(
            false, a, false, bb, (short)0, acc, false, false);
    }

    float xn = x2[(size_t)b * NN + j0 + m];
    size_t cb = (size_t)b * NR * NN;
    #pragma unroll
    for (int r = 0; r < 8; ++r) {
        int row = r + ((lane >> 4) << 3);      // +8 for upper lane half
        float sq = r2[i0 + row] + xn - 2.0f * acc[r];
        C[cb + (size_t)(i0 + row) * NN + j0 + m] = sqrtf(fmaxf(sq, 0.0f));
    }
}

// ---------------------------------------------------------------------------
// Kernel 4: u[b,i] = eps*(log_wx - LSE_j((v[b,j]-C[b,i,j])/eps))
// One wave per row; online (streaming) logsumexp, shfl-xor merge.
// ---------------------------------------------------------------------------
__global__ void u_kernel(const float* __restrict__ C,
                         const float* __restrict__ v,
                         float* __restrict__ u,
                         float inv_eps, float eps, float log_wx) {
    int wid  = (blockIdx.x * blockDim.x + threadIdx.x) >> 5;
    int lane = threadIdx.x & 31;
    int b = wid >> 8;
    int i = wid & 255;
    const float* crow = C + ((size_t)b * NR + i) * NN;
    const float* vrow = v + (size_t)b * NN;

    float m = -3.0e38f, s = 0.0f;
    for (int j = lane; j < NN; j += 32) {
        float x = (vrow[j] - crow[j]) * inv_eps;
        if (x <= m) {
            s += __expf(x - m);
        } else {
            s = s * __expf(m - x) + 1.0f;
            m = x;
        }
    }
    #pragma unroll
    for (int off = 16; off; off >>= 1) {
        float m2 = __shfl_xor(m, off, 32);
        float s2 = __shfl_xor(s, off, 32);
        float M  = fmaxf(m, m2);
        s = s * __expf(m - M) + s2 * __expf(m2 - M);
        m = M;
    }
    if (lane == 0) u[(size_t)b * NR + i] = eps * (log_wx - (m + logf(s)));
}

// ---------------------------------------------------------------------------
// Kernel 5: v[b,j] = eps*(log_wy - LSE_i((u[b,i]-C[b,i,j])/eps))
// One thread per column (coalesced across lanes); u[b,:] staged in LDS.
// ---------------------------------------------------------------------------
__global__ void v_kernel(const float* __restrict__ C,
                         const float* __restrict__ u,
                         float* __restrict__ v,
                         float inv_eps, float eps, float log_wy) {
    __shared__ float us[NR];
    int b  = blockIdx.x >> 3;              // 8 blocks of 256 columns per batch
    int j0 = (blockIdx.x & 7) * 256;
    int t  = threadIdx.x;
    us[t] = u[(size_t)b * NR + t];
    __syncthreads();

    int j = j0 + t;
    const float* ccol = C + (size_t)b * NR * NN + j;
    float m = -3.0e38f, s = 0.0f;
    for (int i = 0; i < NR; ++i) {
        float x = (us[i] - ccol[(size_t)i * NN]) * inv_eps;
        if (x <= m) {
            s += __expf(x - m);
        } else {
            s = s * __expf(m - x) + 1.0f;
            m = x;
        }
    }
    v[(size_t)b * NN + j] = eps * (log_wy - (m + logf(s)));
}

// ---------------------------------------------------------------------------
// Kernel 6: out[b,i,:] = (P[b,i,:] @ X[b]) / (sum_j P + 1e-8) - ref[i,:]
// P computed on the fly; one wave per (b, 16-row tile of i); 8 column tiles
// of d accumulated with WMMA f32 16x16x4; denom via cross-lane shuffles.
// ---------------------------------------------------------------------------
__global__ void final_kernel(const float* __restrict__ C,
                             const float* __restrict__ X,
                             const float* __restrict__ u,
                             const float* __restrict__ v,
                             const float* __restrict__ ref,
                             float* __restrict__ out,
                             float inv_eps) {
    int wid  = (blockIdx.x * blockDim.x + threadIdx.x) >> 5;
    int lane = threadIdx.x & 31;
    int b  = wid >> 4;
    int i0 = (wid & 15) * 16;

    int m     = lane & 15;
    int khalf = (lane >> 4) << 1;          // 0 or 2

    float um = u[(size_t)b * NR + i0 + m];
    const float* crow = C + ((size_t)b * NR + i0 + m) * NN + khalf;
    const float* vrow = v + (size_t)b * NN + khalf;
    const float* Xb   = X + (size_t)b * NN * DD;

    v8f acc[8] = {};
    float dpart = 0.0f;

    for (int k = 0; k < NN; k += 4) {
        v2f a;
        a.x = __expf((um + vrow[k]     - crow[k])     * inv_eps);
        a.y = __expf((um + vrow[k + 1] - crow[k + 1]) * inv_eps);
        dpart += a.x + a.y;
        const float* xr0 = Xb + (size_t)(k + khalf) * DD + m;
        const float* xr1 = xr0 + DD;
        #pragma unroll
        for (int t = 0; t < 8; ++t) {
            v2f bb;
            bb.x = xr0[t * 16];
            bb.y = xr1[t * 16];
            acc[t] = __builtin_amdgcn_wmma_f32_16x16x4_f32(
                false, a, false, bb, (short)0, acc[t], false, false);
        }
    }

    // Row sums of P: lane L holds partial for row L%16; halves merged by xor16.
    float dfull = dpart + __shfl_xor(dpart, 16, 32);

    #pragma unroll
    for (int r = 0; r < 8; ++r) {
        int row = r + ((lane >> 4) << 3);
        float denom = __shfl(dfull, row, 32) + 1e-8f;
        #pragma unroll
        for (int t = 0; t < 8; ++t) {
            int n = t * 16 + m;
            out[((size_t)b * NR + i0 + row) * DD + n] =
                acc[t][r] / denom - ref[(size_t)(i0 + row) * DD + n];
        }
    }
}

// ---------------------------------------------------------------------------
extern "C" void kernel_launch(void* const* d_in, const int* in_sizes, int n_in,
                              void* d_out, int out_size, void* d_ws, size_t ws_size,
                              hipStream_t stream) {
    const float* X   = (const float*)d_in[0];   // (64, 2048, 128)
    const float* ref = (const float*)d_in[1];   // (256, 128)
    float* out = (float*)d_out;                 // (64, 256, 128)

    // Workspace layout (floats): C | u | v | r2 | x2
    float* C  = (float*)d_ws;
    float* u  = C  + (size_t)BB * NR * NN;      // 33,554,432
    float* v  = u  + (size_t)BB * NR;           // + 16,384
    float* r2 = v  + (size_t)BB * NN;           // + 131,072
    float* x2 = r2 + NR;                        // + 256

    const float eps     = 0.001f;
    const float inv_eps = 1.0f / eps;
    const float log_wx  = logf(1.0f / NR + 1e-8f);
    const float log_wy  = logf(1.0f / NN + 1e-8f);

    // 1) squared norms: (256 + 64*2048) rows, one wave each, 8 waves/block
    {
        int rows = NR + BB * NN;                // 131328
        sqnorm_kernel<<<rows / 8, 256, 0, stream>>>(X, ref, r2, x2);
    }
    // 2) v = 0
    zerov_kernel<<<(BB * NN) / 256, 256, 0, stream>>>(v);

    // 3) cost matrix C via WMMA: 64 * 16 * 128 tiles, one wave per tile
    gram_kernel<<<(BB * 16 * 128) / 8, 256, 0, stream>>>(X, ref, r2, x2, C);

    // 4) 100 Sinkhorn iterations (u then v, each a full sweep of L2-resident C)
    for (int it = 0; it < 100; ++it) {
        u_kernel<<<(BB * NR) / 8, 256, 0, stream>>>(C, v, u, inv_eps, eps, log_wx);
        v_kernel<<<(BB * NN) / 256, 256, 0, stream>>>(C, u, v, inv_eps, eps, log_wy);
    }

    // 5) barycenters: P@X via WMMA + fused rowsum/normalize/subtract
    final_kernel<<<(BB * 16) / 8, 256, 0, stream>>>(C, X, u, v, ref, out, inv_eps);
}